// GraphConvolution3_8538394984688
// MI455X (gfx1250) — compile-verified
//
#include <hip/hip_runtime.h>
#include <hip/hip_bf16.h>

// Problem constants from the reference
#define NROWS 8192
#define DIN   64
#define DOUT  64
#define MAXNB 100

typedef __attribute__((ext_vector_type(2))) float v2f;
typedef __attribute__((ext_vector_type(8))) float v8f;

// ---------------------------------------------------------------------------
// Kernel 1: support = input @ weight  (8192x64 @ 64x64, f32)
// One wave per 16x16 output tile, V_WMMA_F32_16X16X4_F32, K-loop over 64 in
// steps of 4. Layouts per CDNA5 ISA 7.12.2:
//   A 16x4:  lanes 0-15 M=0..15 / lanes 16-31 M=0..15; VGPR0={K0|K2}, VGPR1={K1|K3}
//   B 4x16:  lanes hold N; VGPR0={K0|K2}, VGPR1={K1|K3} (mirrored)
//   C/D 16x16 f32: VGPR r, lane<16 -> M=r, lane>=16 -> M=r+8; N = lane%16
// ---------------------------------------------------------------------------
__global__ __launch_bounds__(256) void support_wmma_kernel(
    const float* __restrict__ input,   // [NROWS, DIN]
    const float* __restrict__ weight,  // [DIN, DOUT]
    float* __restrict__ support)       // [NROWS, DOUT]
{
  const int lane = threadIdx.x & 31;
  const int wave = blockIdx.x * (blockDim.x >> 5) + (threadIdx.x >> 5);
  const int mt   = wave >> 2;          // 0..511  (row tile)
  const int nt   = wave & 3;           // 0..3    (col tile)
  const int m0   = mt * 16;
  const int n0   = nt * 16;
  const int half = lane >> 4;          // 0 or 1
  const int l16  = lane & 15;

  v8f c = {};
#pragma unroll
  for (int kk = 0; kk < DIN; kk += 4) {
    // A: row m0+l16, cols kk + 2*half + {0,1}  (contiguous b64 load)
    const float* ap = input + (size_t)(m0 + l16) * DIN + kk + 2 * half;
    v2f a;
    a.x = ap[0];
    a.y = ap[1];
    // B: col n0+l16, rows kk + 2*half + {0,1}
    const float* bp = weight + (size_t)(kk + 2 * half) * DOUT + n0 + l16;
    v2f b;
    b.x = bp[0];
    b.y = bp[DOUT];
    c = __builtin_amdgcn_wmma_f32_16x16x4_f32(
        /*neg_a=*/false, a, /*neg_b=*/false, b,
        /*c_mod=*/(short)0, c, /*reuse_a=*/false, /*reuse_b=*/false);
  }

#pragma unroll
  for (int r = 0; r < 8; ++r) {
    support[(size_t)(m0 + r + 8 * half) * DOUT + n0 + l16] = c[r];
  }
}

// ---------------------------------------------------------------------------
// Kernel 2: out[i] = sum over the last <=100 neighbors j (descending column
// order) of 0.5^r * support[j] + bias, where r = #neighbors after j.
// One wave per row; scan adj row BACKWARDS 32 columns at a time, ballot the
// nonzeros (wave-uniform), walk set bits from high lane to low lane, and
// EARLY-EXIT after 100 neighbors (weights beyond that are exactly zero in the
// reference). Expected columns touched per row ~200 (density 0.5), so adj
// traffic drops from 256 MB to ~8 MB.
// Each lane accumulates features {lane, lane+32}.
// ---------------------------------------------------------------------------
__global__ __launch_bounds__(256) void decay_gather_kernel(
    const float* __restrict__ adj,      // [NROWS, NROWS]
    const float* __restrict__ support,  // [NROWS, DOUT]
    const float* __restrict__ bias,     // [DOUT]
    float* __restrict__ out)            // [NROWS, DOUT]
{
  const int lane = threadIdx.x & 31;
  const int row  = blockIdx.x * (blockDim.x >> 5) + (threadIdx.x >> 5);
  const float* arow = adj + (size_t)row * NROWS;

  float acc0 = 0.0f, acc1 = 0.0f;
  int cnt = 0;

  for (int base = NROWS - 32; base >= 0; base -= 32) {
    if (base >= 128) {
      // backward-stride prefetch hint (emits global_prefetch_b8)
      __builtin_prefetch((const void*)(arow + base - 128 + lane), 0, 1);
    }
    const float v = arow[base + lane];
    unsigned mask = __builtin_amdgcn_ballot_w32(v != 0.0f);
    while (mask) {
      const int l = 31 - __builtin_clz(mask);   // highest set lane = largest column
      mask &= ~(1u << l);
      const float w = __int_as_float((127 - cnt) << 23);  // 2^-cnt (cnt < 100 -> normal)
      const float* sp = support + (size_t)(base + l) * DOUT;
      acc0 = fmaf(w, sp[lane], acc0);
      acc1 = fmaf(w, sp[lane + 32], acc1);
      if (++cnt >= MAXNB) break;  // remaining weights are exactly 0
    }
    if (cnt >= MAXNB) break;
  }

  out[(size_t)row * DOUT + lane]      = acc0 + bias[lane];
  out[(size_t)row * DOUT + lane + 32] = acc1 + bias[lane + 32];
}

extern "C" void kernel_launch(void* const* d_in, const int* in_sizes, int n_in,
                              void* d_out, int out_size, void* d_ws, size_t ws_size,
                              hipStream_t stream) {
  (void)in_sizes; (void)n_in; (void)out_size; (void)ws_size;
  const float* input  = (const float*)d_in[0];  // [8192,64]
  const float* adj    = (const float*)d_in[1];  // [8192,8192]
  const float* weight = (const float*)d_in[2];  // [64,64]
  const float* bias   = (const float*)d_in[3];  // [64]
  float* out     = (float*)d_out;               // [8192,64]
  float* support = (float*)d_ws;                // 8192*64*4 = 2 MB scratch

  // 2048 tiles of 16x16, 8 waves (256 threads) per block -> 256 blocks
  support_wmma_kernel<<<256, 256, 0, stream>>>(input, weight, support);
  // 8192 rows, 1 wave per row, 8 waves per block -> 1024 blocks
  decay_gather_kernel<<<1024, 256, 0, stream>>>(adj, support, bias, out);
}